// WKV_43044162240910
// MI455X (gfx1250) — compile-verified
//
#include <hip/hip_runtime.h>
#include <stdint.h>

// Problem shape (fixed by the reference).
#define BB 8
#define TT 2048
#define DD 1024

constexpr int CH  = 64;        // chunk length along T
constexpr int NC  = TT / CH;   // 32 chunks -> 8*32 = 256 blocks in passes 1/2
constexpr int TPB = 256;       // 256 threads * 4 channels = full D row per block
constexpr int VEC = 4;

// -------- gfx1250 async global->LDS availability (device pass only) --------
#if defined(__HIP_DEVICE_COMPILE__) && defined(__has_builtin)
#  if __has_builtin(__builtin_amdgcn_global_load_async_to_lds_b128)
#    define HAVE_ASYNC_LDS 1
#  endif
#endif
#ifndef HAVE_ASYNC_LDS
#  define HAVE_ASYNC_LDS 0
#endif

// Builtin signature (from hipcc diagnostic): params are pointers to
// 'int __attribute__((vector_size(16)))' — global src (AS1), LDS dst (AS3).
typedef int v4i __attribute__((vector_size(16)));
typedef __attribute__((address_space(1))) v4i* gptr_v4i;
typedef __attribute__((address_space(3))) v4i* lptr_v4i;

__device__ __forceinline__ gptr_v4i to_gas(const void* p) {
  return (gptr_v4i)(uintptr_t)p;            // generic->global: same bits
}
__device__ __forceinline__ lptr_v4i to_lds(const void* p) {
  return (lptr_v4i)(uint32_t)(uintptr_t)p;  // low 32 bits = LDS offset
}

// ---------------------------------------------------------------------------
// Pass 1: per-chunk affine contributions from zero state.
//   C_A = sum_{t in chunk} ed^{(end-1-t)} * inc_t * e^{k_t} v_t
//   C_B = sum_{t in chunk} eg^{(end-1-t)} * inc_t * e^{-k_t}
// computed as the recurrence a = ed*a + f_t (identical fp ordering to ref).
// ---------------------------------------------------------------------------
__global__ void __launch_bounds__(TPB, 1)
wkv_pass1(const float* __restrict__ k, const float* __restrict__ v,
          const float* __restrict__ w,
          float* __restrict__ CA, float* __restrict__ CB) {
  const int b = blockIdx.x, c = blockIdx.y, tid = threadIdx.x;
  const int d0 = tid * VEC;
  const int t0 = c * CH;

  const float4 w4 = *(const float4*)(w + d0);
  float ed[VEC] = {expf(-w4.x), expf(-w4.y), expf(-w4.z), expf(-w4.w)};
  float eg[VEC] = {expf( w4.x), expf( w4.y), expf( w4.z), expf( w4.w)};

  float a[VEC] = {0.f, 0.f, 0.f, 0.f};
  float bm[VEC] = {0.f, 0.f, 0.f, 0.f};

  const size_t row0 = ((size_t)b * TT + t0) * DD + d0;

#pragma unroll 4
  for (int tt = 0; tt < CH; ++tt) {
    if (tt + 16 < CH) {
      __builtin_prefetch(k + row0 + (size_t)(tt + 16) * DD, 0, 1);
      __builtin_prefetch(v + row0 + (size_t)(tt + 16) * DD, 0, 1);
    }
    const float4 k4 = *(const float4*)(k + row0 + (size_t)tt * DD);
    const float4 v4 = *(const float4*)(v + row0 + (size_t)tt * DD);
    const int t = t0 + tt;
    const float inc = (t == 0) ? 0.0f : 1.0f;  // i=0 never contributes
    const float kk[VEC] = {k4.x, k4.y, k4.z, k4.w};
    const float vv[VEC] = {v4.x, v4.y, v4.z, v4.w};
#pragma unroll
    for (int i = 0; i < VEC; ++i) {
      a[i]  = ed[i] * a[i]  + inc * expf(kk[i]) * vv[i];
      bm[i] = eg[i] * bm[i] + inc * expf(-kk[i]);
    }
  }

  const size_t sidx = ((size_t)b * NC + c) * DD + d0;
  *(float4*)(CA + sidx) = make_float4(a[0], a[1], a[2], a[3]);
  *(float4*)(CB + sidx) = make_float4(bm[0], bm[1], bm[2], bm[3]);
}

// ---------------------------------------------------------------------------
// Pass 1.5: exclusive scan over chunks (NC=32) per (b,d) -> start states,
// written in place over the contribution arrays.
// ---------------------------------------------------------------------------
__global__ void wkv_scan(const float* __restrict__ w,
                         float* __restrict__ CA, float* __restrict__ CB) {
  const int gid = blockIdx.x * blockDim.x + threadIdx.x;
  if (gid >= BB * (DD / VEC)) return;
  const int b  = gid / (DD / VEC);
  const int d0 = (gid % (DD / VEC)) * VEC;

  const float4 w4 = *(const float4*)(w + d0);
  const float wv[VEC] = {w4.x, w4.y, w4.z, w4.w};
  float edC[VEC], egC[VEC];
#pragma unroll
  for (int i = 0; i < VEC; ++i) {
    edC[i] = expf(-wv[i] * (float)CH);
    egC[i] = expf( wv[i] * (float)CH);
  }

  float a[VEC] = {0.f, 0.f, 0.f, 0.f};
  float bm[VEC] = {0.f, 0.f, 0.f, 0.f};
  for (int c = 0; c < NC; ++c) {
    const size_t idx = ((size_t)b * NC + c) * DD + d0;
    const float4 ca4 = *(const float4*)(CA + idx);
    const float4 cb4 = *(const float4*)(CB + idx);
    *(float4*)(CA + idx) = make_float4(a[0], a[1], a[2], a[3]);
    *(float4*)(CB + idx) = make_float4(bm[0], bm[1], bm[2], bm[3]);
    const float ca[VEC] = {ca4.x, ca4.y, ca4.z, ca4.w};
    const float cb[VEC] = {cb4.x, cb4.y, cb4.z, cb4.w};
#pragma unroll
    for (int i = 0; i < VEC; ++i) {
      a[i]  = edC[i] * a[i]  + ca[i];
      bm[i] = egC[i] * bm[i] + cb[i];
    }
  }
}

// ---------------------------------------------------------------------------
// Pass 2: outputs. Streams k/v rows through LDS with a DEPTH-deep async
// global->LDS pipeline (gfx1250 ASYNCcnt path); falls back to direct
// global_load_b128 + global_prefetch_b8 if the builtin is unavailable.
// ---------------------------------------------------------------------------
__global__ void __launch_bounds__(TPB, 1)
wkv_pass2(const float* __restrict__ k, const float* __restrict__ v,
          const float* __restrict__ w, const float* __restrict__ u,
          const float* __restrict__ SA, const float* __restrict__ SB,
          float* __restrict__ out) {
  const int b = blockIdx.x, c = blockIdx.y, tid = threadIdx.x;
  const int d0 = tid * VEC;
  const int t0 = c * CH;

  const float4 w4 = *(const float4*)(w + d0);
  const float4 u4 = *(const float4*)(u + d0);
  const float uu[VEC] = {u4.x, u4.y, u4.z, u4.w};
  float ed[VEC] = {expf(-w4.x), expf(-w4.y), expf(-w4.z), expf(-w4.w)};
  float eg[VEC] = {expf( w4.x), expf( w4.y), expf( w4.z), expf( w4.w)};

  const size_t sidx = ((size_t)b * NC + c) * DD + d0;
  const float4 a4 = *(const float4*)(SA + sidx);
  const float4 b4 = *(const float4*)(SB + sidx);
  float a[VEC]  = {a4.x, a4.y, a4.z, a4.w};
  float bm[VEC] = {b4.x, b4.y, b4.z, b4.w};

  const size_t row0 = ((size_t)b * TT + t0) * DD + d0;

  auto body = [&](int tt, float4 k4, float4 v4) {
    const int t = t0 + tt;
    const float tm1 = (t >= 1) ? (float)(t - 1) : 0.0f;  // max(t-1, 0)
    const float inc = (t == 0) ? 0.0f : 1.0f;
    const float kk[VEC] = {k4.x, k4.y, k4.z, k4.w};
    const float vv[VEC] = {v4.x, v4.y, v4.z, v4.w};
    float o[VEC];
#pragma unroll
    for (int i = 0; i < VEC; ++i) {
      const float eku = expf(uu[i] + kk[i]);
      const float ek  = expf(kk[i]);
      const float emk = expf(-kk[i]);
      o[i]  = a[i] + eku * (vv[i] * bm[i] + tm1);
      a[i]  = ed[i] * a[i]  + inc * ek * vv[i];
      bm[i] = eg[i] * bm[i] + inc * emk;
    }
    *(float4*)(out + row0 + (size_t)tt * DD) = make_float4(o[0], o[1], o[2], o[3]);
  };

#if HAVE_ASYNC_LDS
  constexpr int DEPTH = 8;                 // 16 b128 async loads in flight/thread
  constexpr int SLOT  = TPB * 2 * VEC;     // floats per slot: k row | v row
  __shared__ __align__(16) float lds[DEPTH * SLOT];  // 64 KB
  float* kslot0 = lds + tid * VEC;
  float* vslot0 = lds + TPB * VEC + tid * VEC;

  auto issue = [&](int tt, int slot) {
    __builtin_amdgcn_global_load_async_to_lds_b128(
        to_gas(k + row0 + (size_t)tt * DD), to_lds(kslot0 + slot * SLOT), 0, 0);
    __builtin_amdgcn_global_load_async_to_lds_b128(
        to_gas(v + row0 + (size_t)tt * DD), to_lds(vslot0 + slot * SLOT), 0, 0);
  };

#pragma unroll
  for (int i = 0; i < DEPTH; ++i) issue(i, i);

  int tt = 0;
  for (; tt < CH - DEPTH; ++tt) {
    // 2*DEPTH issued; <=14 outstanding => the oldest pair (this slot) landed.
    asm volatile("s_wait_asynccnt 0xe" ::: "memory");
    const int slot = tt & (DEPTH - 1);
    const float4 k4 = *(const float4*)(kslot0 + slot * SLOT);
    const float4 v4 = *(const float4*)(vslot0 + slot * SLOT);
    // Drain DS reads before the async engine may overwrite this slot.
    asm volatile("s_wait_dscnt 0x0" ::: "memory");
    issue(tt + DEPTH, slot);
    body(tt, k4, v4);
  }
  asm volatile("s_wait_asynccnt 0x0" ::: "memory");
  for (; tt < CH; ++tt) {
    const int slot = tt & (DEPTH - 1);
    const float4 k4 = *(const float4*)(kslot0 + slot * SLOT);
    const float4 v4 = *(const float4*)(vslot0 + slot * SLOT);
    body(tt, k4, v4);
  }
#else
#pragma unroll 4
  for (int tt = 0; tt < CH; ++tt) {
    if (tt + 16 < CH) {
      __builtin_prefetch(k + row0 + (size_t)(tt + 16) * DD, 0, 1);
      __builtin_prefetch(v + row0 + (size_t)(tt + 16) * DD, 0, 1);
    }
    const float4 k4 = *(const float4*)(k + row0 + (size_t)tt * DD);
    const float4 v4 = *(const float4*)(v + row0 + (size_t)tt * DD);
    body(tt, k4, v4);
  }
#endif
}

// ---------------------------------------------------------------------------
extern "C" void kernel_launch(void* const* d_in, const int* in_sizes, int n_in,
                              void* d_out, int out_size, void* d_ws, size_t ws_size,
                              hipStream_t stream) {
  (void)in_sizes; (void)n_in; (void)out_size; (void)ws_size;
  const float* k = (const float*)d_in[0];
  const float* v = (const float*)d_in[1];
  const float* w = (const float*)d_in[2];
  const float* u = (const float*)d_in[3];
  float* out = (float*)d_out;

  float* CA = (float*)d_ws;                      // [B, NC, D] chunk sums -> starts
  float* CB = CA + (size_t)BB * NC * DD;         // 2 * 1 MiB of workspace

  dim3 grid(BB, NC);
  wkv_pass1<<<grid, TPB, 0, stream>>>(k, v, w, CA, CB);
  const int nscan = BB * (DD / VEC);
  wkv_scan<<<(nscan + TPB - 1) / TPB, TPB, 0, stream>>>(w, CA, CB);
  wkv_pass2<<<grid, TPB, 0, stream>>>(k, v, w, u, CA, CB, out);
}